// MultiHeadedAttention2D_17540646436958
// MI455X (gfx1250) — compile-verified
//
#include <hip/hip_runtime.h>
#include <math.h>

// MultiHeadedAttention2D for MI455X (gfx1250, wave32, WMMA).
// All matmuls via v_wmma_f32_16x16x4_f32 (fp32 in, fp32 acc) — reference is fp32.

typedef __attribute__((ext_vector_type(2))) float v2f;
typedef __attribute__((ext_vector_type(8))) float v8f;

static constexpr int Bsz = 16;
static constexpr int D   = 512;
static constexpr int L   = 4096;
static constexpr int H   = 8;
static constexpr int Lh  = L / H;   // 512

__device__ __forceinline__ v8f wmma_f32(v2f a, v2f b, v8f c) {
    return __builtin_amdgcn_wmma_f32_16x16x4_f32(false, a, false, b, (short)0, c,
                                                 false, false);
}

// ---------------------------------------------------------------------------
// Projection: Y[b,o,l] = sum_i W[o,i] * X[b,i,l] + bias[o]
// Grid: (L/256, D/64, B). Block: 256 threads = 8 waves as 2(M) x 4(N).
// Wave tile: 32(M=o) x 64(N=l), K = D stepped by 4 -> 8 WMMAs / k-step,
// each B operand reused by 2 WMMAs (10.7 FLOP/B of operand traffic).
// ---------------------------------------------------------------------------
__global__ __launch_bounds__(256) void proj_kernel(const float* __restrict__ X,
                                                   const float* __restrict__ W,
                                                   const float* __restrict__ bias,
                                                   float* __restrict__ Y) {
    const int lane = threadIdx.x & 31;
    const int wave = threadIdx.x >> 5;
    const int lm   = lane & 15;
    const int half = lane >> 4;          // 0: lanes 0-15, 1: lanes 16-31

    const int wm = wave >> 2;            // 0..1
    const int wn = wave & 3;             // 0..3

    const int b     = blockIdx.z;
    const int oBase = blockIdx.y * 64 + wm * 32;
    const int lBase = blockIdx.x * 256 + wn * 64;

    // A(m,k) = W[(oBase+m)*D + k]; per-lane: row oBase+lm (+16), k = kb + 2*half + {0,1}
    const float* __restrict__ A0 = W + (size_t)(oBase + lm) * D + 2 * half;
    const float* __restrict__ A1 = A0 + (size_t)16 * D;
    // B(k,n) = X[b*D*L + k*L + lBase + n]; per-lane n = nt*16 + lm
    const float* __restrict__ Bptr = X + (size_t)b * D * L + (size_t)(2 * half) * L
                                       + lBase + lm;

    v8f c00 = {}, c01 = {}, c02 = {}, c03 = {};
    v8f c10 = {}, c11 = {}, c12 = {}, c13 = {};

    for (int kb = 0; kb < D; kb += 4) {
        const v2f a0 = *(const v2f*)(A0 + kb);
        const v2f a1 = *(const v2f*)(A1 + kb);
        const float* bp = Bptr + (size_t)kb * L;
        v2f b0, b1, b2, b3;
        b0.x = bp[0];      b0.y = bp[L];
        b1.x = bp[16];     b1.y = bp[L + 16];
        b2.x = bp[32];     b2.y = bp[L + 32];
        b3.x = bp[48];     b3.y = bp[L + 48];
        c00 = wmma_f32(a0, b0, c00);
        c10 = wmma_f32(a1, b0, c10);
        c01 = wmma_f32(a0, b1, c01);
        c11 = wmma_f32(a1, b1, c11);
        c02 = wmma_f32(a0, b2, c02);
        c12 = wmma_f32(a1, b2, c12);
        c03 = wmma_f32(a0, b3, c03);
        c13 = wmma_f32(a1, b3, c13);
    }

    // C layout: vgpr r -> m = r + 8*half, n = lm
    float* __restrict__ Yb = Y + (size_t)b * D * L + lBase + lm;
#pragma unroll
    for (int r = 0; r < 8; ++r) {
        const int m0 = r + 8 * half;          // rows oBase + m0
        const int m1 = m0 + 16;               // rows oBase + 16 + m0
        const float bv0 = bias[oBase + m0];
        const float bv1 = bias[oBase + m1];
        float* yr0 = Yb + (size_t)(oBase + m0) * L;
        float* yr1 = Yb + (size_t)(oBase + m1) * L;
        yr0[0]  = c00[r] + bv0;
        yr0[16] = c01[r] + bv0;
        yr0[32] = c02[r] + bv0;
        yr0[48] = c03[r] + bv0;
        yr1[0]  = c10[r] + bv1;
        yr1[16] = c11[r] + bv1;
        yr1[32] = c12[r] + bv1;
        yr1[48] = c13[r] + bv1;
    }
}

// ---------------------------------------------------------------------------
// Fused attention for one (h, b, 16-row d-tile):
//   S = scale * Qh[d:d+16,:] * Kh^T  (16 x 512, into LDS)
//   P = softmax_rows(S)
//   X = P * Vh                        (16 x 512, to global)
// Grid: (D/16, B, H). Block: 256 threads = 8 waves; each wave owns 64 N-cols.
// ---------------------------------------------------------------------------
__global__ __launch_bounds__(256) void attn_kernel(const float* __restrict__ q,
                                                   const float* __restrict__ k,
                                                   const float* __restrict__ v,
                                                   float* __restrict__ x) {
    __shared__ float sP[16 * Lh];        // 32 KB score/probability stripe
    __shared__ float red[256];
    __shared__ float rowv[16];

    const int lane = threadIdx.x & 31;
    const int wave = threadIdx.x >> 5;
    const int lm   = lane & 15;
    const int half = lane >> 4;

    const int dBase = blockIdx.x * 16;
    const int b     = blockIdx.y;
    const int h     = blockIdx.z;
    const int nBase = wave * 64;         // this wave's 64 columns of N=512

    const size_t bRow = (size_t)b * D;   // row index base into (B*D, L)
    const float scale = 0.044194173824159216f;  // 1/sqrt(512)

    // ---- Phase 1: S(16 x 512) = Qh * Kh^T -------------------------------
    {
        // A(m,kk) = q[(bRow + dBase + m)*L + h*Lh + kk]  (contiguous kk pair)
        const float* __restrict__ Ap =
            q + (bRow + dBase + lm) * L + h * Lh + 2 * half;
        // B(kk,n) = k[(bRow + nBase + n)*L + h*Lh + kk]  (contiguous kk pair)
        const float* __restrict__ B0 =
            k + (bRow + nBase + lm) * L + h * Lh + 2 * half;

        v8f acc0 = {}, acc1 = {}, acc2 = {}, acc3 = {};
        for (int kb = 0; kb < Lh; kb += 4) {
            const v2f a  = *(const v2f*)(Ap + kb);
            const v2f b0 = *(const v2f*)(B0 + kb);
            const v2f b1 = *(const v2f*)(B0 + (size_t)16 * L + kb);
            const v2f b2 = *(const v2f*)(B0 + (size_t)32 * L + kb);
            const v2f b3 = *(const v2f*)(B0 + (size_t)48 * L + kb);
            acc0 = wmma_f32(a, b0, acc0);
            acc1 = wmma_f32(a, b1, acc1);
            acc2 = wmma_f32(a, b2, acc2);
            acc3 = wmma_f32(a, b3, acc3);
        }
#pragma unroll
        for (int r = 0; r < 8; ++r) {
            const int m = r + 8 * half;
            sP[m * Lh + nBase + lm]      = acc0[r] * scale;
            sP[m * Lh + nBase + 16 + lm] = acc1[r] * scale;
            sP[m * Lh + nBase + 32 + lm] = acc2[r] * scale;
            sP[m * Lh + nBase + 48 + lm] = acc3[r] * scale;
        }
    }
    __syncthreads();

    // ---- Phase 2: row softmax over 512 columns (16 threads per row) -----
    {
        const int row = threadIdx.x >> 4;    // 0..15
        const int sub = threadIdx.x & 15;    // 0..15
        float mx = -INFINITY;
        for (int j = sub; j < Lh; j += 16) mx = fmaxf(mx, sP[row * Lh + j]);
        red[row * 16 + sub] = mx;
        __syncthreads();
        if (sub == 0) {
            float m2 = red[row * 16];
#pragma unroll
            for (int t = 1; t < 16; ++t) m2 = fmaxf(m2, red[row * 16 + t]);
            rowv[row] = m2;
        }
        __syncthreads();
        const float rmax = rowv[row];
        float s = 0.0f;
        for (int j = sub; j < Lh; j += 16) {
            const float e = __expf(sP[row * Lh + j] - rmax);
            sP[row * Lh + j] = e;
            s += e;
        }
        __syncthreads();                      // rowv fully read before reuse
        red[row * 16 + sub] = s;
        __syncthreads();
        if (sub == 0) {
            float t2 = 0.0f;
#pragma unroll
            for (int t = 0; t < 16; ++t) t2 += red[row * 16 + t];
            rowv[row] = 1.0f / t2;
        }
        __syncthreads();
        const float inv = rowv[row];
        for (int j = sub; j < Lh; j += 16) sP[row * Lh + j] *= inv;
    }
    __syncthreads();

    // ---- Phase 3: X(16 x 512) = P * Vh ----------------------------------
    {
        // A(m,kk) = sP[m*Lh + kk] (LDS, ds_load_b64 per lane)
        const float* __restrict__ Ap = &sP[lm * Lh + 2 * half];
        // B(kk,n) = v[(bRow + kk)*L + h*Lh + nBase + n]
        const float* __restrict__ Bp =
            v + (bRow + 2 * half) * L + h * Lh + nBase + lm;

        v8f acc0 = {}, acc1 = {}, acc2 = {}, acc3 = {};
        for (int kb = 0; kb < Lh; kb += 4) {
            const v2f a = *(const v2f*)(Ap + kb);
            const float* bp = Bp + (size_t)kb * L;
            v2f b0, b1, b2, b3;
            b0.x = bp[0];   b0.y = bp[L];
            b1.x = bp[16];  b1.y = bp[L + 16];
            b2.x = bp[32];  b2.y = bp[L + 32];
            b3.x = bp[48];  b3.y = bp[L + 48];
            acc0 = wmma_f32(a, b0, acc0);
            acc1 = wmma_f32(a, b1, acc1);
            acc2 = wmma_f32(a, b2, acc2);
            acc3 = wmma_f32(a, b3, acc3);
        }

        float* __restrict__ Xp = x + (bRow + dBase) * L + h * Lh + nBase + lm;
#pragma unroll
        for (int r = 0; r < 8; ++r) {
            const int m = r + 8 * half;
            float* xr = Xp + (size_t)m * L;
            xr[0]  = acc0[r];
            xr[16] = acc1[r];
            xr[32] = acc2[r];
            xr[48] = acc3[r];
        }
    }
}

// ---------------------------------------------------------------------------
extern "C" void kernel_launch(void* const* d_in, const int* in_sizes, int n_in,
                              void* d_out, int out_size, void* d_ws, size_t ws_size,
                              hipStream_t stream) {
    const float* xq = (const float*)d_in[0];
    const float* xk = (const float*)d_in[1];
    const float* xv = (const float*)d_in[2];
    const float* Wq = (const float*)d_in[3];
    const float* bq = (const float*)d_in[4];
    const float* Wk = (const float*)d_in[5];
    const float* bk = (const float*)d_in[6];
    const float* Wv = (const float*)d_in[7];
    const float* bv = (const float*)d_in[8];
    const float* Wo = (const float*)d_in[9];
    const float* bo = (const float*)d_in[10];
    float* out = (float*)d_out;

    const size_t N = (size_t)Bsz * D * L;   // 33,554,432 floats each
    float* qBuf = (float*)d_ws;
    float* kBuf = qBuf + N;
    float* vBuf = kBuf + N;
    float* xBuf = vBuf + N;                 // total 4N floats = 512 MiB of ws

    dim3 pgrid(L / 256, D / 64, Bsz);       // (16,8,16)
    dim3 agrid(D / 16, Bsz, H);             // (32,16,8)

    proj_kernel<<<pgrid, 256, 0, stream>>>(xq, Wq, bq, qBuf);
    proj_kernel<<<pgrid, 256, 0, stream>>>(xk, Wk, bk, kBuf);
    proj_kernel<<<pgrid, 256, 0, stream>>>(xv, Wv, bv, vBuf);
    attn_kernel<<<agrid, 256, 0, stream>>>(qBuf, kBuf, vBuf, xBuf);
    proj_kernel<<<pgrid, 256, 0, stream>>>(xBuf, Wo, bo, out);
}